// PairwiseMarginRankingLoss_25975962206323
// MI455X (gfx1250) — compile-verified
//
#include <hip/hip_runtime.h>

// CDNA5 / gfx1250, wave32.
typedef __attribute__((ext_vector_type(2))) float v2f;
typedef __attribute__((ext_vector_type(8))) float v8f;

#define NDIM   512
#define TILE   16
#define TPD    32          // tiles per dimension (512/16)
#define BLOCK  256         // 8 wave32s
#define WAVES  8
#define TI_PER_WAVE (TPD / WAVES)   // 4 i-tiles per wave
#define BIGNEG (-1.0e30f)

// One block per batch row. Computes
//   rowTotals[r] = sum_{i,j} relu(1 - P_i + N_j)   (masks folded into P/N)
//   rowPairs[r]  = npos * nneg
__global__ __launch_bounds__(BLOCK)
void PairwiseMarginRankingLoss_row_kernel(const float* __restrict__ scores,
                                          const int*   __restrict__ labels,
                                          float*        __restrict__ rowTotals,
                                          unsigned int* __restrict__ rowPairs) {
  __shared__ float sA[NDIM];   // pos ? -score  : -1e30   (A-matrix column, K=0)
  __shared__ float sB[NDIM];   // neg ? 1+score : -1e30   (B-matrix row,    K=1)
  __shared__ int   cnt[2];     // npos, nneg (integer atomics: order-independent)
  __shared__ float red[BLOCK];

  const int r   = blockIdx.x;
  const int tid = threadIdx.x;
  if (tid < 2) cnt[tid] = 0;

  // Only HBM traffic of this block: one scores row + one labels row.
  __builtin_prefetch(&scores[r * NDIM], 0, 0);   // -> global_prefetch_b8
  __builtin_prefetch(&labels[r * NDIM], 0, 0);
  __syncthreads();

  // Stage the row into LDS with masks folded in. Masked entries push the
  // hinge argument to ~-1e30 (or -2e30 when both masked) -> relu = 0 exactly,
  // with no inf/NaN anywhere in the WMMA.
  int locPos = 0, locNeg = 0;
  for (int i = tid; i < NDIM; i += BLOCK) {
    const float s = scores[r * NDIM + i];
    const int   l = labels[r * NDIM + i];
    const bool pos = (l == 1);
    const bool neg = (l == 0);
    sA[i] = pos ? -s         : BIGNEG;
    sB[i] = neg ? (1.0f + s) : BIGNEG;
    locPos += pos ? 1 : 0;
    locNeg += neg ? 1 : 0;
  }
  atomicAdd(&cnt[0], locPos);
  atomicAdd(&cnt[1], locNeg);
  __syncthreads();

  const int lane = tid & 31;       // wave32
  const int wave = tid >> 5;
  const int l16  = lane & 15;
  // Lane-half mask, computed ONCE: lanes 0-15 carry K={0,1}, lanes 16-31
  // (K={2,3}) contribute zero rows/cols. Arithmetic masking keeps every LDS
  // load unconditional -> no exec-save/restore in the hot loop, and EXEC is
  // trivially all-ones at each WMMA as the ISA requires.
  const float hmask = (lane < 16) ? 1.0f : 0.0f;

  // Wave w owns i-tiles {4w .. 4w+3}; A is built once per 32 inner WMMAs.
  float acc0 = 0.0f, acc1 = 0.0f, acc2 = 0.0f, acc3 = 0.0f;
  for (int tii = 0; tii < TI_PER_WAVE; ++tii) {
    const int ti = (wave * TI_PER_WAVE + tii) * TILE;

    v2f a;                               // A(16x4): [-P | 1 | 0 | 0]
    a.x = sA[ti + l16] * hmask;
    a.y = hmask;

    for (int tj = 0; tj < NDIM; tj += TILE) {
      v2f b;                             // B(4x16): [1 ; 1+N ; 0 ; 0]
      b.x = hmask;
      b.y = sB[tj + l16] * hmask;

      v8f c = {};
      // D[i,j] = (-P_i)*1 + 1*(1+N_j) = 1 - P_i + N_j   (exact f32 WMMA)
      v8f d = __builtin_amdgcn_wmma_f32_16x16x4_f32(
          /*neg_a=*/false, a, /*neg_b=*/false, b,
          /*c_mod=*/(short)0, c, /*reuse_a=*/false, /*reuse_b=*/false);

      // Nonlinearity + accumulate on VALU; 4 independent chains for ILP.
      acc0 += fmaxf(d[0], 0.0f);
      acc1 += fmaxf(d[1], 0.0f);
      acc2 += fmaxf(d[2], 0.0f);
      acc3 += fmaxf(d[3], 0.0f);
      acc0 += fmaxf(d[4], 0.0f);
      acc1 += fmaxf(d[5], 0.0f);
      acc2 += fmaxf(d[6], 0.0f);
      acc3 += fmaxf(d[7], 0.0f);
    }
  }
  const float acc = (acc0 + acc1) + (acc2 + acc3);

  // Deterministic fixed-order block tree reduction.
  red[tid] = acc;
  __syncthreads();
  for (int s = BLOCK / 2; s > 0; s >>= 1) {
    if (tid < s) red[tid] += red[tid + s];
    __syncthreads();
  }
  if (tid == 0) {
    rowTotals[r] = red[0];
    rowPairs[r]  = (unsigned int)cnt[0] * (unsigned int)cnt[1];
  }
}

// Final reduction over 512 row partials + the num_pairs>0 guard.
__global__ __launch_bounds__(BLOCK)
void PairwiseMarginRankingLoss_final_kernel(const float* __restrict__ rowTotals,
                                            const unsigned int* __restrict__ rowPairs,
                                            float* __restrict__ out) {
  __shared__ float        fr[BLOCK];
  __shared__ unsigned int pr[BLOCK];
  const int tid = threadIdx.x;
  float        t = 0.0f;
  unsigned int p = 0u;
  for (int i = tid; i < NDIM; i += BLOCK) {
    t += rowTotals[i];
    p += rowPairs[i];
  }
  fr[tid] = t;
  pr[tid] = p;
  __syncthreads();
  for (int s = BLOCK / 2; s > 0; s >>= 1) {
    if (tid < s) { fr[tid] += fr[tid + s]; pr[tid] += pr[tid + s]; }
    __syncthreads();
  }
  if (tid == 0) out[0] = (pr[0] > 0u) ? (fr[0] / (float)pr[0]) : 0.0f;
}

extern "C" void kernel_launch(void* const* d_in, const int* in_sizes, int n_in,
                              void* d_out, int out_size, void* d_ws, size_t ws_size,
                              hipStream_t stream) {
  (void)in_sizes; (void)n_in; (void)out_size; (void)ws_size;
  const float* scores = (const float*)d_in[0];
  const int*   labels = (const int*)d_in[1];

  float*        rowTotals = (float*)d_ws;                                        // 512 f32
  unsigned int* rowPairs  = (unsigned int*)((char*)d_ws + NDIM * sizeof(float)); // 512 u32

  PairwiseMarginRankingLoss_row_kernel<<<NDIM, BLOCK, 0, stream>>>(
      scores, labels, rowTotals, rowPairs);
  PairwiseMarginRankingLoss_final_kernel<<<1, BLOCK, 0, stream>>>(
      rowTotals, rowPairs, (float*)d_out);
}